// TopicModel_17841294147984
// MI455X (gfx1250) — compile-verified
//
#include <hip/hip_runtime.h>
#include <math.h>

// Problem constants (from reference)
#define VV   50000
#define DD   300
#define KK   50
#define BB   32
#define SS   8192
#define NN   128
#define LL   64

typedef __attribute__((ext_vector_type(2))) float v2f;
typedef __attribute__((ext_vector_type(8))) float v8f;

// ---------------------------------------------------------------------------
// Kernel A: fused per-segment pipeline.
// One workgroup (256 threads = 8 waves) per segment. Gather masked embedding
// tile E[64,300] into LDS ONCE, then:
//   y = masked mean (write out)            [B,N,D]
//   v = M_w^T y   (per-segment matvec)
//   esc_l = E_l . v, masked softmax -> alpha
//   z = sum_l alpha_l E_l (write out)      [B,N,D]
// ---------------------------------------------------------------------------
__global__ void seg_attn_kernel(const float* __restrict__ emb,
                                const float* __restrict__ Mw,
                                const int*   __restrict__ tokens,
                                const int*   __restrict__ starts,
                                const int*   __restrict__ lens,
                                float* __restrict__ out_y,
                                float* __restrict__ out_z) {
  extern __shared__ float smem[];
  float* sE     = smem;                 // LL*DD
  float* sy     = sE + LL * DD;         // 304
  float* sv     = sy + 304;             // 304
  float* sesc   = sv + 304;             // 64
  float* salpha = sesc + LL;            // 64
  int*   sTok   = (int*)(salpha + LL);  // 64

  const int seg  = blockIdx.x;          // 0..B*N-1
  const int b    = seg / NN;
  const int tid  = threadIdx.x;
  const int start = starts[seg];
  const int len   = lens[seg];

  if (tid < LL) sTok[tid] = tokens[b * SS + start + tid];
  __syncthreads();

  // Gather masked embedding rows (rows l >= len are zero, matching reference).
  for (int idx = tid; idx < LL * DD; idx += blockDim.x) {
    const int l = idx / DD;
    const int d = idx - l * DD;
    sE[idx] = (l < len) ? emb[(long)sTok[l] * DD + d] : 0.0f;
  }
  __syncthreads();

  // y = masked mean over tokens
  const float invLen = 1.0f / fmaxf((float)len, 1.0f);
  for (int d = tid; d < DD; d += blockDim.x) {
    float acc = 0.0f;
    for (int l = 0; l < LL; ++l) acc += sE[l * DD + d];
    const float yv = acc * invLen;
    sy[d] = yv;
    out_y[(long)seg * DD + d] = yv;
  }
  __syncthreads();

  // v[d] = sum_e M_w[e,d] * y[e]   (coalesced M_w reads across threads)
  for (int d = tid; d < DD; d += blockDim.x) {
    float acc = 0.0f;
    for (int e = 0; e < DD; ++e) acc += Mw[(long)e * DD + d] * sy[e];
    sv[d] = acc;
  }
  __syncthreads();

  // esc_l = E_l . v  (masked)
  if (tid < LL) {
    float acc = 0.0f;
    for (int d = 0; d < DD; ++d) acc += sE[tid * DD + d] * sv[d];
    sesc[tid] = (tid < len) ? acc : -1e9f;
  }
  __syncthreads();

  // softmax stats (computed redundantly per thread from LDS)
  float mx = -1e30f;
  for (int l = 0; l < LL; ++l) mx = fmaxf(mx, sesc[l]);
  float ssum = 0.0f;
  for (int l = 0; l < LL; ++l) ssum += expf(sesc[l] - mx);
  if (tid < LL) salpha[tid] = expf(sesc[tid] - mx) / ssum;
  __syncthreads();

  // z = alpha^T E
  for (int d = tid; d < DD; d += blockDim.x) {
    float acc = 0.0f;
    for (int l = 0; l < LL; ++l) acc += salpha[l] * sE[l * DD + d];
    out_z[(long)seg * DD + d] = acc;
  }
}

// ---------------------------------------------------------------------------
// Kernel B: logits = Z[4096,300] @ W_w^T   -> Lbuf [4096 x 64]
// f32 WMMA 16x16x4. One wave per 16x16 tile; 4 waves/block cover N=0..63.
// A layout (16x4 f32): lanes 0-15 row M, VGPR0/1 = K0/K1; lanes 16-31 = K2/K3.
// B layout symmetric over N. C/D: VGPR j -> row j (lanes<16) / j+8 (lanes>=16).
// Columns n>=50 use a CLAMPED address (no exec-mask branching): the resulting
// garbage stays confined to pad columns of Lbuf which are never read.
// ---------------------------------------------------------------------------
__global__ void logits_wmma_kernel(const float* __restrict__ Z,
                                   const float* __restrict__ Ww,
                                   float* __restrict__ Lbuf) {
  const int lane = threadIdx.x & 31;
  const int wave = threadIdx.x >> 5;
  const int M0   = blockIdx.x * 16;
  const int N0   = wave * 16;             // 0,16,32,48
  const int half = lane >> 4;             // 0 or 1
  const int mrow = M0 + (lane & 15);
  const int n    = N0 + (lane & 15);
  const int nn   = (n < KK) ? n : 0;      // clamp: keep loads in-bounds

  const float* Arow = Z  + (long)mrow * DD;
  const float* Brow = Ww + (long)nn   * DD;

  v8f c = {};
  for (int k0 = 0; k0 < DD; k0 += 4) {    // 300/4 = 75 WMMA steps
    const int ka = k0 + 2 * half;
    v2f a  = *(const v2f*)(Arow + ka);    // global_load_b64
    v2f bv = *(const v2f*)(Brow + ka);    // global_load_b64, unconditional
    c = __builtin_amdgcn_wmma_f32_16x16x4_f32(false, a, false, bv,
                                              (short)0, c, false, false);
  }
  const int rbase = M0 + half * 8;
  #pragma unroll
  for (int j = 0; j < 8; ++j)
    Lbuf[(long)(rbase + j) * 64 + n] = c[j];
}

// ---------------------------------------------------------------------------
// Kernel C: row softmax over K=50 logits -> p (out, [B*N,50]) and zero-padded
// Pbuf [B*N x 64] for the next WMMA GEMM.
// ---------------------------------------------------------------------------
__global__ void softmax_p_kernel(const float* __restrict__ Lbuf,
                                 float* __restrict__ out_p,
                                 float* __restrict__ Pbuf) {
  const int row = blockIdx.x * blockDim.x + threadIdx.x;
  if (row >= BB * NN) return;
  const float* lr = Lbuf + (long)row * 64;
  float mx = -1e30f;
  for (int k = 0; k < KK; ++k) mx = fmaxf(mx, lr[k]);
  float ssum = 0.0f;
  for (int k = 0; k < KK; ++k) ssum += expf(lr[k] - mx);
  const float inv = 1.0f / ssum;
  for (int k = 0; k < 64; ++k) {
    const float pv = (k < KK) ? expf(lr[k] - mx) * inv : 0.0f;
    Pbuf[(long)row * 64 + k] = pv;
    if (k < KK) out_p[(long)row * KK + k] = pv;
  }
}

// ---------------------------------------------------------------------------
// Kernel D: r = P[4096,50] @ T_w^T  -> [4096,300].  B[k,d] = T_w[d*50+k].
// K padded to 52 (13 WMMA steps). A pad columns (k>=50) are exact zeros in
// Pbuf, so for the tail pair the B address is clamped to k=0: the loaded
// finite values multiply zeros and contribute nothing. N-edge (n>=300) is
// clamped on load and guarded on store (after WMMA, EXEC all-1s during WMMA).
// ---------------------------------------------------------------------------
__global__ void r_wmma_kernel(const float* __restrict__ Pbuf,
                              const float* __restrict__ Tw,
                              float* __restrict__ out_r) {
  const int lane = threadIdx.x & 31;
  const int wave = threadIdx.x >> 5;
  const int M0   = blockIdx.x * 16;
  const int N0   = (blockIdx.y * 4 + wave) * 16;
  if (N0 >= DD) return;                  // wave-uniform: whole wave exits
  const int half = lane >> 4;
  const int mrow = M0 + (lane & 15);
  const int n    = N0 + (lane & 15);
  const int nn   = (n < DD) ? n : 0;     // clamp for in-bounds loads

  const float* Arow = Pbuf + (long)mrow * 64;
  const float* Brow = Tw   + (long)nn   * KK;

  v8f c = {};
  for (int k0 = 0; k0 < 52; k0 += 4) {   // 13 WMMA steps cover K=50
    const int ka = k0 + 2 * half;
    const int kb = (ka + 1 < KK) ? ka : 0;      // clamp tail pair (A there = 0)
    v2f a  = *(const v2f*)(Arow + ka);          // zero-padded beyond 49
    v2f bv = *(const v2f*)(Brow + kb);          // unconditional b64 load
    c = __builtin_amdgcn_wmma_f32_16x16x4_f32(false, a, false, bv,
                                              (short)0, c, false, false);
  }
  if (n < DD) {
    const int rbase = M0 + half * 8;
    #pragma unroll
    for (int j = 0; j < 8; ++j)
      out_r[(long)(rbase + j) * DD + n] = c[j];
  }
}

// ---------------------------------------------------------------------------
extern "C" void kernel_launch(void* const* d_in, const int* in_sizes, int n_in,
                              void* d_out, int out_size, void* d_ws, size_t ws_size,
                              hipStream_t stream) {
  const float* emb    = (const float*)d_in[0];  // [V,D]
  const float* Mw     = (const float*)d_in[1];  // [D,D]
  const float* Ww     = (const float*)d_in[2];  // [K,D]
  const float* Tw     = (const float*)d_in[3];  // [D,K]
  const int*   tokens = (const int*)  d_in[4];  // [B,S]
  const int*   starts = (const int*)  d_in[5];  // [B,N]
  const int*   lens   = (const int*)  d_in[6];  // [B,N]

  float* out   = (float*)d_out;
  const long BND = (long)BB * NN * DD;
  float* out_y = out;
  float* out_z = out + BND;
  float* out_p = out + 2 * BND;
  float* out_r = out_p + (long)BB * NN * KK;

  float* Lbuf = (float*)d_ws;                   // [B*N x 64] padded logits
  float* Pbuf = Lbuf + (long)BB * NN * 64;      // [B*N x 64] padded p

  // Kernel A: 4096 segments, 8 waves each, ~80 KB dynamic LDS per workgroup.
  const size_t ldsBytes =
      (size_t)(LL * DD + 304 + 304 + LL + LL + LL) * sizeof(float);
  seg_attn_kernel<<<BB * NN, 256, ldsBytes, stream>>>(
      emb, Mw, tokens, starts, lens, out_y, out_z);

  // Kernel B: 256 row-tiles x 4 N-tiles (N padded to 64).
  logits_wmma_kernel<<<BB * NN / 16, 128, 0, stream>>>(out_z, Ww, Lbuf);

  // Kernel C: softmax over K=50 per row.
  softmax_p_kernel<<<(BB * NN + 255) / 256, 256, 0, stream>>>(Lbuf, out_p, Pbuf);

  // Kernel D: 256 row-tiles x 19 N-tiles (grid.y=5, 4 waves each, tail guarded).
  dim3 gridR(BB * NN / 16, 5);
  r_wmma_kernel<<<gridR, 128, 0, stream>>>(Pbuf, Tw, out_r);
}